// TransposeCASSI_38439957299669
// MI455X (gfx1250) — compile-verified
//
#include <hip/hip_runtime.h>

// Problem constants from the reference (B=4, M=512, L=31, S=2, W=M+L-1).
#define MDIM 512
#define LDIM 31
#define WDIM 542                  // M + L - 1
#define ROWE (MDIM * LDIM)        // 15872 elements per (b,m) block
#define NV4  (ROWE / 4)           // 3968 float4 per (b,m)
#define YROW (WDIM * 2)           // 1084 floats of Y per (b,m)
#define YCHUNKS ((YROW * 4) / 16) // 271 b128 chunks
#define TPB  256

typedef float v4f __attribute__((ext_vector_type(4)));

// Build ysum[w] = Y[w,0] + Y[w,1] in LDS. On gfx1250, stage the raw Y row
// into LDS with async global->LDS b128 transfers (ASYNCcnt path), then reduce.
__device__ __forceinline__ void build_ysum(const float* __restrict__ yrow,
                                           float* __restrict__ ysum,
                                           float* __restrict__ yraw,
                                           int tid)
{
#if defined(__gfx1250__)
    for (int c = tid; c < YCHUNKS; c += TPB) {
        const float* gp = yrow + 4 * c;
        unsigned la = (unsigned)(size_t)(yraw + 4 * c);   // LDS byte offset
        asm volatile("global_load_async_to_lds_b128 %0, %1, off"
                     :: "v"(la), "v"(gp) : "memory");
    }
    asm volatile("s_wait_asynccnt 0x0" ::: "memory");
    __syncthreads();
    for (int w = tid; w < WDIM; w += TPB)
        ysum[w] = yraw[2 * w] + yraw[2 * w + 1];
#else
    (void)yraw;
    for (int w = tid; w < WDIM; w += TPB)
        ysum[w] = yrow[2 * w] + yrow[2 * w + 1];
#endif
    __syncthreads();
}

__global__ void k_init(unsigned* gmax) { *gmax = 0u; }

// Pass 1: compute products, track global max (no stores of X).
// H is read with default (RT) temporal hint so it stays resident in the
// 192 MB L2 for pass 2 (H is 130 MB).
__global__ void __launch_bounds__(TPB)
k_max(const float* __restrict__ Y, const float* __restrict__ H,
      unsigned* __restrict__ gmax)
{
    __shared__ float ysum[WDIM];
    __shared__ float yraw[YROW];
    const int bm  = blockIdx.x;
    const int tid = threadIdx.x;

    build_ysum(Y + (size_t)bm * YROW, ysum, yraw, tid);

    const v4f* __restrict__ H4 = (const v4f*)(H + (size_t)bm * ROWE);
    float vmax = 0.0f;
    for (int i = tid; i < NV4; i += TPB) {
        __builtin_prefetch(H4 + i + TPB, 0, 0);   // gfx1250 global_prefetch
        v4f h = H4[i];
#pragma unroll
        for (int j = 0; j < 4; ++j) {
            unsigned e = 4u * (unsigned)i + (unsigned)j;
            unsigned r = e / 31u;          // magic-multiply divide
            unsigned w = e - 30u * r;      // w = r + l = e - 30*r
            float v = h[j] * ysum[w];
            vmax = fmaxf(vmax, v);
        }
    }

    // wave32 max reduction, then one atomic per wave (values >= 0, so
    // uint-bit compare == float compare).
#pragma unroll
    for (int off = 16; off >= 1; off >>= 1)
        vmax = fmaxf(vmax, __shfl_xor(vmax, off, 32));
    if ((tid & 31) == 0)
        atomicMax(gmax, __float_as_uint(vmax));
}

// Pass 2: recompute products (H hot in L2) and scale by 1/max.
// X stores are non-temporal so the output stream does not evict H from L2.
__global__ void __launch_bounds__(TPB)
k_out(const float* __restrict__ Y, const float* __restrict__ H,
      float* __restrict__ X, const unsigned* __restrict__ gmax)
{
    __shared__ float ysum[WDIM];
    __shared__ float yraw[YROW];
    const int bm  = blockIdx.x;
    const int tid = threadIdx.x;

    const float inv = 1.0f / __uint_as_float(*gmax);   // one IEEE divide, hoisted
    build_ysum(Y + (size_t)bm * YROW, ysum, yraw, tid);

    const v4f* __restrict__ H4 = (const v4f*)(H + (size_t)bm * ROWE);
    v4f* __restrict__ X4 = (v4f*)(X + (size_t)bm * ROWE);
    for (int i = tid; i < NV4; i += TPB) {
        __builtin_prefetch(H4 + i + TPB, 0, 0);
        v4f h = H4[i];
        v4f o;
#pragma unroll
        for (int j = 0; j < 4; ++j) {
            unsigned e = 4u * (unsigned)i + (unsigned)j;
            unsigned r = e / 31u;
            unsigned w = e - 30u * r;
            o[j] = (h[j] * ysum[w]) * inv;
        }
        __builtin_nontemporal_store(o, X4 + i);        // th:TH_STORE_NT
    }
}

extern "C" void kernel_launch(void* const* d_in, const int* in_sizes, int n_in,
                              void* d_out, int out_size, void* d_ws, size_t ws_size,
                              hipStream_t stream)
{
    (void)n_in; (void)out_size; (void)ws_size;
    const float* Y = (const float*)d_in[0];   // (B, M, W, S) float32
    const float* H = (const float*)d_in[1];   // (B, M, M, L) float32
    float* X       = (float*)d_out;           // (B, M, M, L) float32
    unsigned* gmax = (unsigned*)d_ws;         // 4 bytes of scratch

    const int BM = in_sizes[1] / ROWE;        // B*M = 2048 blocks

    k_init<<<1, 1, 0, stream>>>(gmax);
    k_max<<<BM, TPB, 0, stream>>>(Y, H, gmax);
    k_out<<<BM, TPB, 0, stream>>>(Y, H, X, gmax);
}